// SeedGNNT_3504693313914
// MI455X (gfx1250) — compile-verified
//
#include <hip/hip_runtime.h>
#include <math.h>

#define D 64
#define WT_STRIDE 66   // pad 64->66 floats: keeps v2f LDS reads 8B-aligned, spreads banks

typedef float v2f __attribute__((ext_vector_type(2)));
typedef float v8f __attribute__((ext_vector_type(8)));

// ---------------------------------------------------------------- utilities
__global__ void fill_zero_kernel(float* __restrict__ p, long long n) {
    long long i = (long long)blockIdx.x * blockDim.x + threadIdx.x;
    long long stride = (long long)gridDim.x * blockDim.x;
    for (; i < n; i += stride) p[i] = 0.0f;
}

// ------------------------------------------------------- edge normalization
__global__ void degree_kernel(const int* __restrict__ etype,
                              const int* __restrict__ col,
                              float* __restrict__ deg0, float* __restrict__ deg1,
                              int nE) {
    int e = blockIdx.x * blockDim.x + threadIdx.x;
    if (e >= nE) return;
    int c = col[e];
    if (etype[e] == 0) atomicAdd(&deg0[c], 1.0f);
    else               atomicAdd(&deg1[c], 1.0f);
}

__global__ void dn_kernel(float* __restrict__ d, int n) {
    int i = blockIdx.x * blockDim.x + threadIdx.x;
    if (i >= n) return;
    float v = d[i];
    d[i] = (v > 0.0f) ? rsqrtf(v) : 0.0f;
}

__global__ void value_kernel(const int* __restrict__ etype,
                             const int* __restrict__ row,
                             const int* __restrict__ col,
                             const float* __restrict__ w,
                             const float* __restrict__ dn0,
                             const float* __restrict__ dn1,
                             float* __restrict__ v0, float* __restrict__ v1,
                             int nE) {
    int e = blockIdx.x * blockDim.x + threadIdx.x;
    if (e >= nE) return;
    int r = row[e], c = col[e];
    float wv = w[e];
    int t = etype[e];
    v0[e] = (t == 0) ? wv * dn0[c] * dn0[r] : 0.0f;
    v1[e] = (t == 1) ? wv * dn1[c] * dn1[r] : 0.0f;
}

// ------------------------------------------------------------------ SpMM
// agg[col] += value[e] * h[row]; 4 threads per edge, 16 channels each.
// Gather via float4 (global_load_b128), scatter via f32 atomics.
__global__ void spmm_scatter_kernel(const int* __restrict__ row,
                                    const int* __restrict__ col,
                                    const float* __restrict__ val,
                                    const float* __restrict__ H,
                                    float* __restrict__ Agg, int nE) {
    int t = blockIdx.x * blockDim.x + threadIdx.x;
    int e = t >> 2;
    if (e >= nE) return;
    float v = val[e];
    if (v == 0.0f) return;
    int cpart = (t & 3) * 16;
    const float4* hs = (const float4*)(H + (size_t)row[e] * D + cpart);
    float4 h0 = hs[0], h1 = hs[1], h2 = hs[2], h3 = hs[3];
    float* ad = Agg + (size_t)col[e] * D + cpart;
    atomicAdd(ad + 0,  v * h0.x); atomicAdd(ad + 1,  v * h0.y);
    atomicAdd(ad + 2,  v * h0.z); atomicAdd(ad + 3,  v * h0.w);
    atomicAdd(ad + 4,  v * h1.x); atomicAdd(ad + 5,  v * h1.y);
    atomicAdd(ad + 6,  v * h1.z); atomicAdd(ad + 7,  v * h1.w);
    atomicAdd(ad + 8,  v * h2.x); atomicAdd(ad + 9,  v * h2.y);
    atomicAdd(ad + 10, v * h2.z); atomicAdd(ad + 11, v * h2.w);
    atomicAdd(ad + 12, v * h3.x); atomicAdd(ad + 13, v * h3.y);
    atomicAdd(ad + 14, v * h3.z); atomicAdd(ad + 15, v * h3.w);
}

// --------------------------------------------------- fused GEMM + BN + ReLU
// Hout = relu( (X @ W + bias) * (bng/sqrt(1+eps)) + bnb ) [+ res]
// X:[nrows,64], W:[64,64] row-major. W is staged TRANSPOSED into LDS once per
// block (coalesced b128 reads), then each V_WMMA_F32_16X16X4_F32 B-operand is
// one aligned ds_load_b64. One wave per 16-row strip; 4 column tiles.
__global__ __launch_bounds__(256) void gemm64_bn_relu_kernel(
    const float* __restrict__ X, const float* __restrict__ W,
    const float* __restrict__ bias, const float* __restrict__ bng,
    const float* __restrict__ bnb, const float* __restrict__ res,
    float* __restrict__ Hout, int nrows) {
    __shared__ float wt[D * WT_STRIDE];      // wt[j*WT_STRIDE + i] = W[i*D + j]

    // ---- cooperative transpose-stage of W (all 256 threads, before barrier)
    {
        const float4* w4 = (const float4*)W;
#pragma unroll
        for (int q = 0; q < 4; ++q) {
            int p = threadIdx.x * 16 + q * 4;      // flat index into W, 4-aligned
            float4 wv = w4[p >> 2];
            int i = p >> 6, j = p & 63;            // p..p+3 share i (never cross row)
            wt[(j + 0) * WT_STRIDE + i] = wv.x;
            wt[(j + 1) * WT_STRIDE + i] = wv.y;
            wt[(j + 2) * WT_STRIDE + i] = wv.z;
            wt[(j + 3) * WT_STRIDE + i] = wv.w;
        }
    }
    __syncthreads();

    const int lane  = threadIdx.x & 31;
    const int wave  = threadIdx.x >> 5;
    const int row0  = (blockIdx.x * 8 + wave) * 16;
    if (row0 >= nrows) return;               // wave-uniform; EXEC all-ones below
    const int mrow  = lane & 15;             // M row (A) / N column (B,C,D)
    const int khalf = lane >> 4;             // K pair select

    v8f acc[4] = {};
    const float* xrow = X + (size_t)(row0 + mrow) * D + 2 * khalf;
#pragma unroll
    for (int k = 0; k < D; k += 4) {
        v2f a;
        a.x = xrow[k];
        a.y = xrow[k + 1];
        const int i0 = k + 2 * khalf;        // even -> 8B-aligned LDS reads
#pragma unroll
        for (int ct = 0; ct < 4; ++ct) {
            v2f b = *(const v2f*)&wt[(ct * 16 + mrow) * WT_STRIDE + i0];
            acc[ct] = __builtin_amdgcn_wmma_f32_16x16x4_f32(
                false, a, false, b, (short)0, acc[ct], false, false);
        }
    }

    const float inv = 0.9999950000374997f;   // 1/sqrt(1+1e-5)
#pragma unroll
    for (int ct = 0; ct < 4; ++ct) {
        int c = ct * 16 + mrow;
        float sc = bng[c] * inv, sh = bnb[c], bs = bias[c];
#pragma unroll
        for (int r = 0; r < 8; ++r) {
            int rowi = row0 + khalf * 8 + r; // D layout: VGPR r -> M=r / r+8
            float v = (acc[ct][r] + bs) * sc + sh;
            v = fmaxf(v, 0.0f);
            if (res) v += res[(size_t)rowi * D + c];
            Hout[(size_t)rowi * D + c] = v;
        }
    }
}

// ------------------------------------------------- seed-attention constant
// softmax over singleton axis == 1, so each attention layer adds a constant
// vector; only dot(seed_emb, out_W[2D:3D]) + out_b reaches the output.
__global__ void seed_kernel(const float* __restrict__ x, const int* __restrict__ sid,
                            const float* Wv0, const float* bv0, const float* Wo0,
                            const float* bo0, const float* g0, const float* b0,
                            const float* Wv1, const float* bv1, const float* Wo1,
                            const float* bo1, const float* g1, const float* b1,
                            const float* __restrict__ outW,
                            const float* __restrict__ outb,
                            float* __restrict__ seedc) {
    __shared__ float sf[D], tv[D], to[D];
    const float inv = 0.9999950000374997f;
    int j = threadIdx.x;
    sf[j] = x[(size_t)sid[0] * D + j];
    __syncthreads();
    float a = bv0[j];
    for (int i = 0; i < D; ++i) a += sf[i] * Wv0[i * D + j];
    tv[j] = a; __syncthreads();
    a = bo0[j];
    for (int i = 0; i < D; ++i) a += tv[i] * Wo0[i * D + j];
    a = fmaxf(a * (g0[j] * inv) + b0[j], 0.0f);
    __syncthreads();
    sf[j] += a; __syncthreads();
    a = bv1[j];
    for (int i = 0; i < D; ++i) a += sf[i] * Wv1[i * D + j];
    tv[j] = a; __syncthreads();
    a = bo1[j];
    for (int i = 0; i < D; ++i) a += tv[i] * Wo1[i * D + j];
    a = fmaxf(a * (g1[j] * inv) + b1[j], 0.0f);
    __syncthreads();
    sf[j] += a; __syncthreads();
    to[j] = sf[j] * outW[2 * D + j];
    __syncthreads();
    if (j == 0) {
        float s = outb[0];
        for (int i = 0; i < D; ++i) s += to[i];
        seedc[0] = s;
    }
}

// ------------------------------------------------------------- output dots
__global__ void out_first_kernel(const float* __restrict__ G,
                                 const float* __restrict__ outW,
                                 float* __restrict__ out, int n) {
    int i = blockIdx.x * blockDim.x + threadIdx.x;
    if (i >= n) return;
    float s = 0.0f;
#pragma unroll
    for (int c = 0; c < D; ++c) s += G[(size_t)i * D + c] * outW[c];
    out[i] = s;
}

__global__ void out_second_kernel(const float* __restrict__ G,
                                  const float* __restrict__ outW,
                                  const float* __restrict__ seedc,
                                  float* __restrict__ out, int n) {
    int i = blockIdx.x * blockDim.x + threadIdx.x;
    if (i >= n) return;
    float s = 0.0f;
#pragma unroll
    for (int c = 0; c < D; ++c) s += G[(size_t)i * D + c] * outW[D + c];
    out[i] = out[i] + s + seedc[0];
}

// ------------------------------------------------------------------ driver
extern "C" void kernel_launch(void* const* d_in, const int* in_sizes, int n_in,
                              void* d_out, int out_size, void* d_ws, size_t ws_size,
                              hipStream_t stream) {
    const float* x     = (const float*)d_in[0];
    const int*   eidx  = (const int*)d_in[1];
    const int*   etype = (const int*)d_in[2];
    const float* ew    = (const float*)d_in[3];
    const int*   seed  = (const int*)d_in[4];
    // params flattened in setup_inputs() dict insertion order:
    // attn[0]: 5..14 (Wq,bq,Wk,bk,Wv,bv,Wo,bo,bn_g,bn_b); attn[1]: 15..24
    // g0: 25 fc_W, 26 fc_b, 27 bn0_g, 28 bn0_b, 29..36 layers
    // g1: 37..48 ; 49 out_W [3D,1]; 50 out_b [1]
    const int N  = in_sizes[0] / D;
    const int E  = in_sizes[2];
    const int* row = eidx;
    const int* col = eidx + E;
    const float* outW = (const float*)d_in[49];
    const float* outb = (const float*)d_in[50];

    float* ws    = (float*)d_ws;
    float* dn0   = ws;                         // N (degree -> 1/sqrt)
    float* dn1   = dn0 + N;                    // N
    float* v0    = dn1 + N;                    // E
    float* v1    = v0 + E;                     // E
    float* hbuf  = v1 + E;                     // N*D
    float* tbuf  = hbuf + (size_t)N * D;       // N*D
    float* agg   = tbuf + (size_t)N * D;       // N*D
    float* seedc = agg + (size_t)N * D;        // 1

    const int TB = 256;
    const int gE   = (E + TB - 1) / TB;
    const int gN   = (N + TB - 1) / TB;
    const int gNE4 = (4 * E + TB - 1) / TB;
    const long long nd = (long long)N * D;
    const int rowTiles  = (N + 15) / 16;
    const int gemmGrid  = (rowTiles + 7) / 8;

    // edge normalization (typed)
    fill_zero_kernel<<<2048, TB, 0, stream>>>(dn0, 2LL * N);
    degree_kernel<<<gE, TB, 0, stream>>>(etype, col, dn0, dn1, E);
    dn_kernel<<<(2 * N + TB - 1) / TB, TB, 0, stream>>>(dn0, 2 * N);
    value_kernel<<<gE, TB, 0, stream>>>(etype, row, col, ew, dn0, dn1, v0, v1, E);

    // seed-attention constant (scalar)
    seed_kernel<<<1, D, 0, stream>>>(x, seed,
        (const float*)d_in[9],  (const float*)d_in[10], (const float*)d_in[11],
        (const float*)d_in[12], (const float*)d_in[13], (const float*)d_in[14],
        (const float*)d_in[19], (const float*)d_in[20], (const float*)d_in[21],
        (const float*)d_in[22], (const float*)d_in[23], (const float*)d_in[24],
        outW, outb, seedc);

    for (int g = 0; g < 2; ++g) {
        const int B = (g == 0) ? 25 : 37;
        const float* val = (g == 0) ? v0 : v1;
        // h = relu(bn(x @ fc_W + fc_b))
        gemm64_bn_relu_kernel<<<gemmGrid, TB, 0, stream>>>(
            x, (const float*)d_in[B], (const float*)d_in[B + 1],
            (const float*)d_in[B + 2], (const float*)d_in[B + 3],
            nullptr, hbuf, N);
        float* hc = hbuf;
        float* hn = tbuf;
        for (int l = 0; l < 2; ++l) {
            const int L = B + 4 + 4 * l;
            fill_zero_kernel<<<4096, TB, 0, stream>>>(agg, nd);
            spmm_scatter_kernel<<<gNE4, TB, 0, stream>>>(row, col, val, hc, agg, E);
            gemm64_bn_relu_kernel<<<gemmGrid, TB, 0, stream>>>(
                agg, (const float*)d_in[L], (const float*)d_in[L + 1],
                (const float*)d_in[L + 2], (const float*)d_in[L + 3],
                hc, hn, N);
            float* t = hc; hc = hn; hn = t;
        }
        if (g == 0)
            out_first_kernel<<<gN, TB, 0, stream>>>(hc, outW, (float*)d_out, N);
        else
            out_second_kernel<<<gN, TB, 0, stream>>>(hc, outW, seedc, (float*)d_out, N);
    }
}